// YoloLoss_5325759447314
// MI455X (gfx1250) — compile-verified
//
#include <hip/hip_runtime.h>

// YOLO loss: streaming reduction over 2 x (16384,7,7,30) f32 -> scalar.
// Memory-bound: ~193 MB read / 23.3 TB/s  =>  ~8.3 us floor. No matmul
// structure => no WMMA; optimize the data-movement path (b128 NT loads,
// speculative prefetch, wave32 + LDS reduction) instead.

#define LAMBDA_COORD 5.0f
#define LAMBDA_NOOBJ 0.5f
#define NBLK 1024
#define NTHR 256

typedef float v2f __attribute__((ext_vector_type(2)));

__device__ __forceinline__ float iou_ref(float x1, float y1, float w1, float h1,
                                         float x2, float y2, float w2, float h2) {
  // Mirrors the reference exactly (including unconditional inter/union math).
  float xl = fmaxf(x1 - 0.5f * w1, x2 - 0.5f * w2);
  float yt = fmaxf(y1 - 0.5f * h1, y2 - 0.5f * h2);
  float xr = fminf(x1 + 0.5f * w1, x2 + 0.5f * w2);
  float yb = fminf(y1 + 0.5f * h1, y2 + 0.5f * h2);
  float inter = (xr - xl) * (yb - yt);
  float uni   = w1 * h1 + w2 * h2 - inter;
  float safe  = (uni == 0.0f) ? 1.0f : uni;
  float r     = inter / safe;
  return ((xr >= xl) && (yb >= yt)) ? r : 0.0f;
}

// wave32 (CDNA5) + LDS block reduction. Result valid in lane 0 of wave 0.
__device__ __forceinline__ float block_reduce(float v, float* smem) {
  #pragma unroll
  for (int off = 16; off > 0; off >>= 1)
    v += __shfl_down(v, off, 32);          // wave32: width MUST be 32 on gfx1250
  const int lane = threadIdx.x & 31;
  const int wave = threadIdx.x >> 5;
  if (lane == 0) smem[wave] = v;
  __syncthreads();
  if (wave == 0) {
    v = (lane < (NTHR / 32)) ? smem[lane] : 0.0f;
    #pragma unroll
    for (int off = 4; off > 0; off >>= 1)  // NTHR/32 == 8 partials
      v += __shfl_down(v, off, 32);
  }
  return v;
}

__global__ __launch_bounds__(NTHR)
void yolo_partial(const float* __restrict__ pred, const float* __restrict__ targ,
                  float* __restrict__ partial, int ncells) {
  const int tid    = blockIdx.x * NTHR + threadIdx.x;
  const int stride = gridDim.x * NTHR;
  float acc = 0.0f;

  for (int cell = tid; cell < ncells; cell += stride) {
    const v2f* p2 = (const v2f*)(pred + (size_t)cell * 30);  // 120B cell, 8B aligned
    const v2f* t2 = (const v2f*)(targ + (size_t)cell * 30);

    // Unguarded speculative prefetch of next grid-stride cell (~30MB ahead;
    // lives in 192MB L2 until consumed).
    __builtin_prefetch(pred + (size_t)(cell + stride) * 30, 0, 0);
    __builtin_prefetch(targ + (size_t)(cell + stride) * 30, 0, 0);

    float p[30], t[30];
    #pragma unroll
    for (int i = 0; i < 15; ++i) {
      v2f a = __builtin_nontemporal_load(p2 + i);  // stream-once: TH=NT (b128-merged)
      v2f b = __builtin_nontemporal_load(t2 + i);
      p[2 * i] = a.x; p[2 * i + 1] = a.y;
      t[2 * i] = b.x; t[2 * i + 1] = b.y;
    }

    const float t4    = t[4];
    const float obj   = (t4 > 0.0f) ? 1.0f : 0.0f;
    const float noobj = (t4 == 0.0f) ? 1.0f : 0.0f;

    // class loss (channels 10..29), 4 independent FMA chains for ILP
    float c0 = 0.0f, c1 = 0.0f, c2 = 0.0f, c3 = 0.0f;
    #pragma unroll
    for (int k = 10; k < 30; k += 4) {
      float d0 = t[k + 0] - p[k + 0]; c0 = fmaf(d0, d0, c0);
      float d1 = t[k + 1] - p[k + 1]; c1 = fmaf(d1, d1, c1);
      float d2 = t[k + 2] - p[k + 2]; c2 = fmaf(d2, d2, c2);
      float d3 = t[k + 3] - p[k + 3]; c3 = fmaf(d3, d3, c3);
    }
    const float cls = (c0 + c1) + (c2 + c3);

    // no-object confidence loss
    const float d4      = t4 - p[4];
    const float conf_no = d4 * d4;

    // responsible-box selection via IoU
    const float i1 = iou_ref(t[0], t[1], t[2], t[3], p[0], p[1], p[2], p[3]);
    const float i2 = iou_ref(t[0], t[1], t[2], t[3], p[5], p[6], p[7], p[8]);
    const float r1 = (i1 > i2) ? 1.0f : 0.0f;
    const float m1 = obj * r1;
    const float m2 = obj * (1.0f - r1);

    const float e1 = i1 - p[4];
    const float e2 = i2 - p[9];
    const float conf_obj = m1 * e1 * e1 + m2 * e2 * e2;

    const float dx1 = t[0] - p[0], dy1 = t[1] - p[1];
    const float dx2 = t[0] - p[5], dy2 = t[1] - p[6];
    const float xy = m1 * (dx1 * dx1 + dy1 * dy1) + m2 * (dx2 * dx2 + dy2 * dy2);

    const float dw1 = t[2] - p[2], dh1 = t[3] - p[3];
    const float dw2 = t[2] - p[7], dh2 = t[3] - p[8];
    const float wh = m1 * (dw1 * dw1 + dh1 * dh1) + m2 * (dw2 * dw2 + dh2 * dh2);

    acc += LAMBDA_COORD * (xy + wh) + conf_obj + LAMBDA_NOOBJ * conf_no + obj * cls;
  }

  __shared__ float smem[NTHR / 32];
  const float bsum = block_reduce(acc, smem);
  if (threadIdx.x == 0) partial[blockIdx.x] = bsum;  // written by every block, every call
}

__global__ __launch_bounds__(NTHR)
void yolo_final(const float* __restrict__ partial, float* __restrict__ out,
                int n, float inv_batch) {
  float acc = 0.0f;
  for (int i = threadIdx.x; i < n; i += NTHR) acc += partial[i];
  __shared__ float smem[NTHR / 32];
  const float total = block_reduce(acc, smem);
  if (threadIdx.x == 0) out[0] = total * inv_batch;  // overwrites poisoned d_out
}

extern "C" void kernel_launch(void* const* d_in, const int* in_sizes, int n_in,
                              void* d_out, int out_size, void* d_ws, size_t ws_size,
                              hipStream_t stream) {
  (void)n_in; (void)out_size; (void)ws_size;
  const float* pred = (const float*)d_in[0];  // y
  const float* targ = (const float*)d_in[1];  // gt
  float* out        = (float*)d_out;
  float* partial    = (float*)d_ws;           // NBLK floats of scratch

  const int ncells = in_sizes[0] / 30;        // B*S*S
  const int batch  = ncells / 49;             // S = 7
  const float inv_b = 1.0f / (float)batch;

  yolo_partial<<<NBLK, NTHR, 0, stream>>>(pred, targ, partial, ncells);
  yolo_final<<<1, NTHR, 0, stream>>>(partial, out, NBLK, inv_b);
}